// JustBalance_22943715295834
// MI455X (gfx1250) — compile-verified
//
#include <hip/hip_runtime.h>
#include <hip/hip_bf16.h>
#include <math.h>

// ---------------------------------------------------------------------------
// JustBalance pooling forward for MI455X (gfx1250, wave32, WMMA).
// Live outputs: s = softmax(h1 @ Wc + bc)  [N,K]   and scalar loss.
// Dense adjacency / out / out_adj in the reference are dead code -> skipped.
// ---------------------------------------------------------------------------

#define NODES 16384
#define INCH  128
#define HID   256
#define KCL   16
#define KC    16          // K-chunk staged in LDS per iteration

typedef float v2f __attribute__((ext_vector_type(2)));
typedef float v8f __attribute__((ext_vector_type(8)));

// ---------------------------------------------------------------------------
// Tiled FP32 WMMA GEMM: out[N x Fout] = A[N x Kd] @ W[Kd x Fout]
// Block: 256 threads (8 waves) -> 64x64 output tile.
// Wave (rw,cw) computes rows rw*16..+16, cols cw*32..+32 (2 accumulators,
// one shared A fragment per K-step). A and W staged through LDS; W is stored
// transposed with pitch 17 so fragment reads are bank-conflict-free
// (17 coprime with the 64 LDS banks).
//
// ISA layouts (CDNA5 ISA):
//   A 16x4 f32 : lane m=L&15; K pair (0,1) lanes 0-15 / (2,3) lanes 16-31
//   B 4x16 f32 : lane n=L&15; same K-pair split
//   C/D 16x16  : VGPR i -> row i + 8*(L>>4), col = L&15
// ---------------------------------------------------------------------------
__global__ __launch_bounds__(256)
void gemm_wmma_tiled(const float* __restrict__ A, const float* __restrict__ W,
                     float* __restrict__ out, int Kd, int Fout) {
    __shared__ float As[64][KC + 1];   // [row][k]
    __shared__ float Wt[64][KC + 1];   // [col][k]  (transposed W chunk)

    const int t    = threadIdx.x;
    const int lane = t & 31;
    const int wave = t >> 5;
    const int rw   = wave & 3;         // row tile within block (0..3)
    const int cw   = wave >> 2;        // column half (0..1)
    const int half = lane >> 4;        // K-pair select
    const int l15  = lane & 15;

    const int rowBase = blockIdx.x * 64;
    const int colBase = blockIdx.y * 64;

    // Cooperative staging assignments (256 threads):
    const int lrow = t >> 2;           // 0..63 : A row within tile
    const int lseg = (t & 3) * 4;      // k offset 0,4,8,12 (float4)
    const int wk   = t >> 4;           // 0..15 : W k-row within chunk
    const int wc   = (t & 15) * 4;     // col offset 0..60 (float4)

    v8f c0 = {}, c1 = {};

    for (int k0 = 0; k0 < Kd; k0 += KC) {
        if (k0 + KC < Kd) {  // global_prefetch_b8 of next A chunk
            __builtin_prefetch(&A[(size_t)(rowBase + lrow) * Kd + k0 + KC + lseg], 0, 0);
        }
        // Stage A[rowBase+lrow][k0+lseg .. +3]
        const float4 av = *(const float4*)&A[(size_t)(rowBase + lrow) * Kd + k0 + lseg];
        As[lrow][lseg + 0] = av.x;  As[lrow][lseg + 1] = av.y;
        As[lrow][lseg + 2] = av.z;  As[lrow][lseg + 3] = av.w;
        // Stage W[k0+wk][colBase+wc .. +3], transposed into Wt[col][k]
        const float4 wv = *(const float4*)&W[(size_t)(k0 + wk) * Fout + colBase + wc];
        Wt[wc + 0][wk] = wv.x;  Wt[wc + 1][wk] = wv.y;
        Wt[wc + 2][wk] = wv.z;  Wt[wc + 3][wk] = wv.w;
        __syncthreads();

#pragma unroll
        for (int kk = 0; kk < KC; kk += 4) {
            const int kb = kk + half * 2;
            v2f a, b0v, b1v;
            a.x   = As[rw * 16 + l15][kb];
            a.y   = As[rw * 16 + l15][kb + 1];
            b0v.x = Wt[cw * 32 + l15][kb];
            b0v.y = Wt[cw * 32 + l15][kb + 1];
            b1v.x = Wt[cw * 32 + 16 + l15][kb];
            b1v.y = Wt[cw * 32 + 16 + l15][kb + 1];
            c0 = __builtin_amdgcn_wmma_f32_16x16x4_f32(
                    false, a, false, b0v, (short)0, c0, false, false);
            c1 = __builtin_amdgcn_wmma_f32_16x16x4_f32(
                    false, a, false, b1v, (short)0, c1, false, false);
        }
        __syncthreads();
    }

    // Write both 16x16 tiles of this wave.
    float* __restrict__ Cp = out + (size_t)(rowBase + rw * 16) * Fout
                                 + colBase + cw * 32 + l15;
#pragma unroll
    for (int i = 0; i < 8; ++i) {
        const int r = i + 8 * half;
        Cp[(size_t)r * Fout]      = c0[i];
        Cp[(size_t)r * Fout + 16] = c1[i];
    }
}

// ---------------------------------------------------------------------------
// Single-wave FP32 WMMA GEMM for the narrow logits matmul (Fout = 16),
// with per-column bias.
// ---------------------------------------------------------------------------
__global__ __launch_bounds__(32)
void gemm_wmma_f32(const float* __restrict__ A, const float* __restrict__ W,
                   const float* __restrict__ bias, float* __restrict__ out,
                   int Kd, int Fout) {
    const int tileM = blockIdx.x;
    const int tileN = blockIdx.y;
    const int lane  = threadIdx.x;
    const int half  = lane >> 4;
    const int l15   = lane & 15;

    const float* __restrict__ Arow = A + (size_t)(tileM * 16 + l15) * Kd;
    const float* __restrict__ Wcol = W + (size_t)(tileN * 16 + l15);

    v8f c = {};
    for (int k = 0; k < Kd; k += 4) {
        const int kb = k + half * 2;
        v2f a, b;
        a.x = Arow[kb];
        a.y = Arow[kb + 1];
        b.x = Wcol[(size_t)kb * Fout];
        b.y = Wcol[(size_t)(kb + 1) * Fout];
        c = __builtin_amdgcn_wmma_f32_16x16x4_f32(
                false, a, false, b, (short)0, c, false, false);
    }

    const float bn = bias ? bias[tileN * 16 + l15] : 0.0f;
    float* __restrict__ Cp = out + (size_t)(tileM * 16) * Fout + tileN * 16 + l15;
#pragma unroll
    for (int i = 0; i < 8; ++i) {
        const int row = i + 8 * half;
        Cp[(size_t)row * Fout] = c[i] + bn;
    }
}

// ---------------------------------------------------------------------------
// Edge aggregation: agg[dst[e], :] += t[src[e], :]   (f32 atomics, L2-resident)
// 32 lanes per edge, lane-interleaved channels -> each atomic instruction
// touches 32 consecutive floats.
// ---------------------------------------------------------------------------
__global__ __launch_bounds__(256)
void scatter_add_edges(const float* __restrict__ t,
                       const int* __restrict__ src, const int* __restrict__ dst,
                       float* __restrict__ agg, int nEdges) {
    const int gid  = blockIdx.x * blockDim.x + threadIdx.x;
    const int e    = gid >> 5;
    const int lane = gid & 31;
    if (e >= nEdges) return;
    const int s = src[e];
    const int d = dst[e];
    const float* __restrict__ trow = t   + (size_t)s * HID;
    float* __restrict__       arow = agg + (size_t)d * HID;
#pragma unroll
    for (int j = 0; j < HID / 32; ++j) {
        const int ch = lane + 32 * j;
        atomicAdd(&arow[ch], trow[ch]);
    }
}

// h = relu(h + b)  (elementwise, bias per channel; HID is a power of two)
__global__ __launch_bounds__(256)
void bias_relu(float* __restrict__ h, const float* __restrict__ b, size_t total) {
    const size_t i = (size_t)blockIdx.x * blockDim.x + threadIdx.x;
    if (i >= total) return;
    const float v = h[i] + b[i & (HID - 1)];
    h[i] = v > 0.0f ? v : 0.0f;
}

// ---------------------------------------------------------------------------
// Row softmax over K=16 + accumulation of colsq[k] = sum_i s[i,k]^2
// (only the diagonal of s^T s feeds the loss).
// ---------------------------------------------------------------------------
__global__ __launch_bounds__(256)
void softmax_colsq(const float* __restrict__ logits, float* __restrict__ s_out,
                   float* __restrict__ colsq) {
    __shared__ float part[KCL];
    const int t = threadIdx.x;
    if (t < KCL) part[t] = 0.0f;
    __syncthreads();

    const int row = blockIdx.x * blockDim.x + t;   // grid covers N exactly
    float v[KCL];
    const float* __restrict__ L = logits + (size_t)row * KCL;
    float mx = -INFINITY;
#pragma unroll
    for (int k = 0; k < KCL; ++k) { v[k] = L[k]; mx = fmaxf(mx, v[k]); }
    float sum = 0.0f;
#pragma unroll
    for (int k = 0; k < KCL; ++k) { v[k] = __expf(v[k] - mx); sum += v[k]; }
    const float inv = 1.0f / sum;
    float* __restrict__ S = s_out + (size_t)row * KCL;
#pragma unroll
    for (int k = 0; k < KCL; ++k) {
        const float sv = v[k] * inv;
        S[k] = sv;
        atomicAdd(&part[k], sv * sv);
    }
    __syncthreads();
    if (t < KCL) atomicAdd(&colsq[t], part[t]);
}

// loss = -sum_k sqrt(colsq[k] + 1e-15) / sqrt(N*K)
__global__ void loss_kernel(const float* __restrict__ colsq, float* __restrict__ out_loss) {
    if (threadIdx.x == 0 && blockIdx.x == 0) {
        float acc = 0.0f;
#pragma unroll
        for (int k = 0; k < KCL; ++k) acc += sqrtf(colsq[k] + 1e-15f);
        out_loss[0] = -acc / sqrtf((float)(NODES * KCL));
    }
}

// ---------------------------------------------------------------------------
// Orchestration. Inputs (setup_inputs order):
//  0:x[N,128] 1:edge_index[2,E] 2:W0[128,256] 3:b0[256] 4:W1[256,256]
//  5:b1[256] 6:Wc[256,16] 7:bc[16]
// Output: s (N*K floats) then loss (1 float).
// Workspace: bufA (N*HID f32), bufB (N*HID f32), colsq (16 f32).
// ---------------------------------------------------------------------------
extern "C" void kernel_launch(void* const* d_in, const int* in_sizes, int n_in,
                              void* d_out, int out_size, void* d_ws, size_t ws_size,
                              hipStream_t stream) {
    const float* x   = (const float*)d_in[0];
    const int*   ei  = (const int*)  d_in[1];   // [2,E]: row0=src, row1=dst
    const float* W0  = (const float*)d_in[2];
    const float* b0  = (const float*)d_in[3];
    const float* W1  = (const float*)d_in[4];
    const float* b1  = (const float*)d_in[5];
    const float* Wc  = (const float*)d_in[6];
    const float* bc  = (const float*)d_in[7];
    float* out = (float*)d_out;

    const int E = in_sizes[1] / 2;
    const int* src = ei;
    const int* dst = ei + E;

    float* bufA  = (float*)d_ws;                        // N*HID
    float* bufB  = bufA + (size_t)NODES * HID;          // N*HID
    float* colsq = bufB + (size_t)NODES * HID;          // 16

    const size_t hidBytes = (size_t)NODES * HID * sizeof(float);
    const dim3 tileGrid(NODES / 64, HID / 64);          // 256 x 4 blocks (64x64 tiles)
    const dim3 gemmGridK(NODES / 16, KCL / 16);         // 1024 x 1 tiles
    const int scatterBlocks = (E * 32) / 256;
    const int ewBlocks = (int)(((size_t)NODES * HID + 255) / 256);

    // Layer 0: t0 = x @ W0 ; agg ; relu(+b0)
    gemm_wmma_tiled<<<tileGrid, 256, 0, stream>>>(x, W0, bufA, INCH, HID);
    hipMemsetAsync(bufB, 0, hidBytes, stream);
    scatter_add_edges<<<scatterBlocks, 256, 0, stream>>>(bufA, src, dst, bufB, E);
    bias_relu<<<ewBlocks, 256, 0, stream>>>(bufB, b0, (size_t)NODES * HID);

    // Layer 1: t1 = h0 @ W1 ; agg ; relu(+b1)
    gemm_wmma_tiled<<<tileGrid, 256, 0, stream>>>(bufB, W1, bufA, HID, HID);
    hipMemsetAsync(bufB, 0, hidBytes, stream);
    scatter_add_edges<<<scatterBlocks, 256, 0, stream>>>(bufA, src, dst, bufB, E);
    bias_relu<<<ewBlocks, 256, 0, stream>>>(bufB, b1, (size_t)NODES * HID);

    // logits = h1 @ Wc + bc  -> bufA[:, :16]
    gemm_wmma_f32<<<gemmGridK, 32, 0, stream>>>(bufB, Wc, bc, bufA, HID, KCL);

    // softmax rows -> d_out[0 : N*K]; colsq diag accumulation; loss -> d_out[N*K]
    hipMemsetAsync(colsq, 0, KCL * sizeof(float), stream);
    softmax_colsq<<<NODES / 256, 256, 0, stream>>>(bufA, out, colsq);
    loss_kernel<<<1, 32, 0, stream>>>(colsq, out + (size_t)NODES * KCL);
}